// RelationAwareTokenizer_12378095747606
// MI455X (gfx1250) — compile-verified
//
#include <hip/hip_runtime.h>

// Problem constants (match reference)
#define N_NODES   50000
#define DIM       128          // floats per node feature
#define NE        600000       // edges
#define NR        500          // relations
#define SEQ       5            // tokens per node
#define NSTEP     4            // SEQ-1 aggregation steps
#define PSTEP     150000       // edges per step
#define TOT_SLOTS (NSTEP * PSTEP)   // 600000

#define BLK_THREADS    1024
#define WAVES_PER_BLK  32
#define EDGES_PER_WAVE 64
#define EDGES_PER_BLK  (WAVES_PER_BLK * EDGES_PER_WAVE)   // 2048

#define REL_F4_TOTAL  (NR * (DIM / 4))      // 16000 float4 = 250 KB
#define TDM_ELEMS     ((NR / 2) * DIM)      // 32000 dwords staged by TDM (125 KB)
#define ASYNC_F4_BEG  (TDM_ELEMS / 4)       // float4 index where async staging starts

typedef unsigned int v4u __attribute__((ext_vector_type(4)));
typedef int          v8i __attribute__((ext_vector_type(8)));
typedef int          v4i __attribute__((ext_vector_type(4)));

// ---------------------------------------------------------------------------
// CDNA5 async global->LDS copy (ASYNCcnt-tracked). VDST = LDS byte address,
// VADDR = 64-bit global address, saddr = off.  (CDNA5 ISA)
// ---------------------------------------------------------------------------
__device__ __forceinline__ void async_load_b128_to_lds(void* lds_dst, const void* gsrc) {
  unsigned           loff = (unsigned)(size_t)lds_dst;   // low 32b of flat LDS addr = LDS offset
  unsigned long long ga   = (unsigned long long)(size_t)gsrc;
  asm volatile("global_load_async_to_lds_b128 %0, %1, off"
               :: "v"(loff), "v"(ga) : "memory");
}

__device__ __forceinline__ void wait_asynccnt0() {
  asm volatile("s_wait_asynccnt 0x0" ::: "memory");
}

// ---------------------------------------------------------------------------
// Kernel A: tokens[:,0,:] = x ; tokens[:,1..4,:] = 0 ; degree ws = 0
// ---------------------------------------------------------------------------
__global__ void init_kernel(const float4* __restrict__ x4,
                            float4* __restrict__ out4,
                            float* __restrict__ deg) {
  int tid = blockIdx.x * blockDim.x + threadIdx.x;
  const int ND4 = N_NODES * (DIM / 4);   // 1.6M float4 chunks
  if (tid < ND4) {
    int n = tid >> 5;        // 32 float4 chunks per node
    int c = tid & 31;
    float4 v = x4[tid];
    float4 z = make_float4(0.f, 0.f, 0.f, 0.f);
    float4* o = out4 + (size_t)(n * SEQ) * 32 + c;   // [n][s][c] in float4 units
    o[0]       = v;   // s = 0 : copy of x
    o[32 * 1]  = z;   // s = 1..4 : zeroed accumulators
    o[32 * 2]  = z;
    o[32 * 3]  = z;
    o[32 * 4]  = z;
  }
  if (tid < NSTEP * N_NODES) deg[tid] = 0.f;
}

// ---------------------------------------------------------------------------
// Kernel B: stage rel_table (250 KB) into LDS — first half via a single TDM
// tensor_load_to_lds descriptor (wave 0), second half via async-to-LDS loads
// (all waves).  The fill streams behind the per-lane edge-metadata phase.
// Then one wave per edge: msg = x[src] + rel_lds[rel]; scatter-add into
// tokens[dst][step+1] and degree.
// ---------------------------------------------------------------------------
__global__ void __launch_bounds__(BLK_THREADS, 1)
scatter_kernel(const float4* __restrict__ x4,
               const float4* __restrict__ rel4,
               const int* __restrict__ eidx,     // [2, E]
               const int* __restrict__ etype,    // [E]
               const int* __restrict__ perms,    // [NSTEP, PSTEP]
               float* __restrict__ out,          // [N, SEQ, DIM]
               float* __restrict__ deg) {        // [NSTEP, N]
  extern __shared__ float4 srel[];               // 16000 float4 = 250 KB

  const int tid  = threadIdx.x;
  const int lane = tid & 31;
  const int wave = tid >> 5;

  // ================= stage: issue =================
#if __has_builtin(__builtin_amdgcn_tensor_load_to_lds)
  if (tid < 32) {   // wave 0 issues one TDM DMA for the first 125 KB
    unsigned           lds_base = (unsigned)(size_t)&srel[0];
    unsigned long long ga       = (unsigned long long)(size_t)rel4;
    // D# group 0: count=1 | lds_addr | global_addr[56:0] | type=2 (image)
    v4u g0 = { 1u,
               lds_base,
               (unsigned)(ga & 0xFFFFFFFFu),
               (unsigned)((ga >> 32) & 0x1FFFFFFu) | 0x80000000u };
    // D# group 1: data_size=2 (4B); tensor_dim0 = tile_dim0 = TDM_ELEMS,
    // tensor_dim1 = tile_dim1 = 1; dim0 stride = TDM_ELEMS
    v8i g1 = { (int)0x00020000u,                             // data_size=4B
               (int)(((unsigned)TDM_ELEMS & 0xFFFFu) << 16), // tensor_dim0[15:0]
               (int)((((unsigned)TDM_ELEMS >> 16) & 0xFFFFu) | (1u << 16)), // dim0 hi | tensor_dim1=1
               (int)(((unsigned)TDM_ELEMS & 0xFFFFu) << 16), // tile_dim0
               1,                                            // tile_dim1 = 1
               (int)TDM_ELEMS,                               // tensor_dim0_stride
               0, 0 };
    v4i gz4 = { 0, 0, 0, 0 };
    v8i gz8 = { 0, 0, 0, 0, 0, 0, 0, 0 };
    __builtin_amdgcn_tensor_load_to_lds(g0, g1, gz4, gz4, gz8, 0);
  }
  // all waves stage the second 125 KB via the async engine
  for (int i = ASYNC_F4_BEG + tid; i < REL_F4_TOTAL; i += BLK_THREADS)
    async_load_b128_to_lds(&srel[i], &rel4[i]);
#else
  for (int i = tid; i < REL_F4_TOTAL; i += BLK_THREADS)
    async_load_b128_to_lds(&srel[i], &rel4[i]);
#endif

  // ============ overlap: per-lane edge metadata + degree (no LDS) ============
  const int base = blockIdx.x * EDGES_PER_BLK + wave * EDGES_PER_WAVE;
  int obase[2], sn[2], rl[2];
#pragma unroll
  for (int g = 0; g < EDGES_PER_WAVE / 32; ++g) {
    const int myslot = base + g * 32 + lane;
    obase[g] = 0; sn[g] = 0; rl[g] = 0;
    if (myslot < TOT_SLOTS) {
      int st = myslot / PSTEP;
      int p  = myslot - st * PSTEP;
      int e  = perms[st * PSTEP + p];
      sn[g]  = eidx[e];            // src node
      int dn = eidx[NE + e];       // dst node
      rl[g]  = etype[e];           // relation id
      obase[g] = (dn * SEQ + st + 1) * DIM;
      atomicAdd(&deg[st * N_NODES + dn], 1.0f);   // degree: 1 atomic per edge
    }
  }

  // ================= stage: drain =================
  wait_asynccnt0();                       // each wave drains its async loads
#if __has_builtin(__builtin_amdgcn_tensor_load_to_lds)
  if (tid < 32) __builtin_amdgcn_s_wait_tensorcnt(0);   // wave 0 drains the TDM
#endif
  __syncthreads();                        // LDS visible to all waves

  // ================= compute: gather + scatter-add =================
#pragma unroll
  for (int g = 0; g < EDGES_PER_WAVE / 32; ++g) {
    const int gbase = base + g * 32;
    int nv = TOT_SLOTS - gbase;
    if (nv > 32) nv = 32;
    for (int j = 0; j < nv; ++j) {
      // broadcast edge j's metadata across the wave32
      int snj = __shfl(sn[g],    j, 32);
      int rlj = __shfl(rl[g],    j, 32);
      int obj = __shfl(obase[g], j, 32);

      float4 xv = x4[(size_t)snj * 32 + lane];   // global_load_b128 (L2-resident x)
      float4 rv = srel[rlj * 32 + lane];         // ds_load_b128 from staged table

      float* o = out + (size_t)obj + lane * 4;
      atomicAdd(o + 0, xv.x + rv.x);
      atomicAdd(o + 1, xv.y + rv.y);
      atomicAdd(o + 2, xv.z + rv.z);
      atomicAdd(o + 3, xv.w + rv.w);
    }
  }
}

// ---------------------------------------------------------------------------
// Kernel C: tokens[n][s+1] /= max(degree[s][n], 1)
// ---------------------------------------------------------------------------
__global__ void norm_kernel(float4* __restrict__ out4,
                            const float* __restrict__ deg) {
  int tid = blockIdx.x * blockDim.x + threadIdx.x;
  const int TOTALT = N_NODES * NSTEP * 32;
  if (tid >= TOTALT) return;
  int c  = tid & 31;
  int t  = tid >> 5;
  int sp = t & 3;          // step 0..3
  int n  = t >> 2;
  float d = fmaxf(deg[sp * N_NODES + n], 1.0f);
  size_t idx = (size_t)(n * SEQ + sp + 1) * 32 + c;
  float4 v = out4[idx];
  v.x /= d; v.y /= d; v.z /= d; v.w /= d;
  out4[idx] = v;
}

// ---------------------------------------------------------------------------
extern "C" void kernel_launch(void* const* d_in, const int* in_sizes, int n_in,
                              void* d_out, int out_size, void* d_ws, size_t ws_size,
                              hipStream_t stream) {
  const float4* x4    = (const float4*)d_in[0];   // [N, 128] fp32
  const float4* rel4  = (const float4*)d_in[1];   // [R, 128] fp32
  const int*    eidx  = (const int*)d_in[2];      // [2, E]
  const int*    etype = (const int*)d_in[3];      // [E]
  const int*    perms = (const int*)d_in[4];      // [4, 150000]
  float*        out   = (float*)d_out;            // [N, 5, 128] fp32
  float*        deg   = (float*)d_ws;             // [4, N] fp32 = 800 KB scratch

  // A: init output + degree
  {
    int total = N_NODES * (DIM / 4);
    int grid  = (total + 255) / 256;
    init_kernel<<<grid, 256, 0, stream>>>(x4, (float4*)out, deg);
  }
  // B: scatter-add with LDS-staged relation table (250 KB dynamic LDS)
  {
    int grid = (TOT_SLOTS + EDGES_PER_BLK - 1) / EDGES_PER_BLK;   // 293 blocks
    size_t lds_bytes = (size_t)NR * DIM * sizeof(float);          // 256000 B
    scatter_kernel<<<grid, BLK_THREADS, lds_bytes, stream>>>(
        x4, rel4, eidx, etype, perms, out, deg);
  }
  // C: normalize by degree
  {
    int total = N_NODES * NSTEP * 32;
    int grid  = (total + 255) / 256;
    norm_kernel<<<grid, 256, 0, stream>>>((float4*)out, deg);
  }
}